// RPNet_19026705122038
// MI455X (gfx1250) — compile-verified
//
#include <hip/hip_runtime.h>
#include <hip/hip_bf16.h>
#include <math.h>

typedef __attribute__((ext_vector_type(2))) float v2f;
typedef __attribute__((ext_vector_type(8))) float v8f;

#define KDIM   53248      // 416 channels * 16 * 8
#define NDIM   128
#define NHEAD  7
#define BATCH  16
#define NCHUNK 32
#define KSTEPS_PER_CHUNK 416   // 53248 / 4 / 32
#define KSTEPS_MAIN      408   // main loop (with stream-ahead prefetch)

// ---------------------------------------------------------------------------
// Kernel 1: ROI adaptive max pool  ->  rois [16][53248]  (row-major, m-major)
//   feature index f = c_global*128 + i*8 + j   (c_global: 0-63 x2, 64-223 x4,
//   224-415 x6), matching reference concat+reshape order.
// ---------------------------------------------------------------------------
__global__ void __launch_bounds__(256)
roi_pool_kernel(const float* __restrict__ box,
                const float* __restrict__ x2,
                const float* __restrict__ x4,
                const float* __restrict__ x6,
                float* __restrict__ rois)
{
    const int TOT = BATCH * 416 * 128;
    int t = blockIdx.x * blockDim.x + threadIdx.x;
    if (t >= TOT) return;

    const int j  = t & 7;          // column bin (out_w = 8)
    const int i  = (t >> 3) & 15;  // row bin    (out_h = 16)
    const int cg = (t >> 7) % 416; // global channel
    const int b  = t / (416 * 128);

    const float* f; int H, C, cl;
    if (cg < 64)       { f = x2; H = 128; C = 64;  cl = cg;        }
    else if (cg < 224) { f = x4; H = 64;  C = 160; cl = cg - 64;   }
    else               { f = x6; H = 32;  C = 192; cl = cg - 224;  }
    const int W = H;   // all levels square (reference scales x by H, y by W)

    const float bx = box[b * 4 + 0];
    const float by = box[b * 4 + 1];
    const float bw = box[b * 4 + 2];
    const float bh = box[b * 4 + 3];

    // box_xyxy = clip(box_xywh @ POSTFIX, 0, 1); then scale [H, W, H, W]
    const float x1f = fminf(fmaxf(bx - 0.5f * bw, 0.0f), 1.0f) * (float)H;
    const float y1f = fminf(fmaxf(by - 0.5f * bh, 0.0f), 1.0f) * (float)W;
    const float x2f = fminf(fmaxf(bx + 0.5f * bw, 0.0f), 1.0f) * (float)H;
    const float y2f = fminf(fmaxf(by + 0.5f * bh, 0.0f), 1.0f) * (float)W;

    // rois.long() truncation (values >= 0, trunc == floor), end clip to dim-1
    const int x1  = (int)x1f;
    const int y1  = (int)y1f;
    const int x2i = min((int)x2f, W - 1);
    const int y2i = min((int)y2f, H - 1);
    const int w   = x2i - x1 + 1;
    const int h   = y2i - y1 + 1;

    // PyTorch adaptive pooling bin boundaries: start=floor, end=ceil
    int cs = x1 + (j * w) / 8;
    int ce = x1 + ((j + 1) * w + 7) / 8;
    int rs = y1 + (i * h) / 16;
    int re = y1 + ((i + 1) * h + 15) / 16;
    ce = min(ce, W);
    re = min(re, H);
    cs = max(cs, 0);
    rs = max(rs, 0);

    const float* fb = f + (((size_t)b * C + cl) * H) * W;
    float v = -INFINITY;
    for (int r = rs; r < re; ++r) {
        const float* row = fb + (size_t)r * W;
        for (int c = cs; c < ce; ++c) v = fmaxf(v, row[c]);
    }

    rois[(size_t)b * KDIM + (size_t)cg * 128 + i * 8 + j] = v;
}

// ---------------------------------------------------------------------------
// Kernel 2: split-K WMMA GEMM:  partial[chunk][head][16][128]
//   D(16x16) = A(16x4,f32) x B(4x16,f32) accumulated over 416 k-steps.
//   One wave per (chunk, head, ntile); 4 waves per block. Two accumulator
//   chains for ILP; all loop addressing via running pointers so the backend
//   folds everything into immediate offsets + one pointer bump per group.
// ---------------------------------------------------------------------------

// one k-step at byte-constant displacement D (in k-steps) from running ptrs
#define GEMM_STEP(ACC, D)                                                      \
    do {                                                                       \
        v2f _a = *(const v2f*)(aPtr + (D) * 4);                                \
        v2f _b;                                                                \
        _b.x = bPtr[(D) * 4 * NDIM];                                           \
        _b.y = bPtr[(D) * 4 * NDIM + NDIM];                                    \
        (ACC) = __builtin_amdgcn_wmma_f32_16x16x4_f32(                         \
            false, _a, false, _b, (short)0, (ACC), false, false);              \
    } while (0)

__global__ void __launch_bounds__(128)
gemm_splitk_wmma(const float* __restrict__ rois,
                 const float* __restrict__ Wh,
                 float* __restrict__ partial)
{
    const int wid  = blockIdx.x * (blockDim.x >> 5) + (threadIdx.x >> 5);
    const int lane = threadIdx.x & 31;

    const int chunk = wid / (NHEAD * 8);         // 0..31
    const int rem   = wid % (NHEAD * 8);
    const int head  = rem >> 3;                  // 0..6
    const int ntile = rem & 7;                   // 0..7  (8 tiles of N=16)

    // ISA A layout (16x4 f32): lanes 0-15 -> M=lane, K=k,k+1 ; lanes 16-31 -> K=k+2,k+3
    const int m    = lane & 15;
    const int koff = (lane >> 4) << 1;
    const int ncol = (ntile << 4) + (lane & 15);

    const int k0 = chunk * (KSTEPS_PER_CHUNK * 4);

    // running pointers (advance by constants; backend uses immediate offsets)
    const float* aPtr = rois + (size_t)m * KDIM + k0 + koff;
    const float* bPtr = Wh + (size_t)head * KDIM * NDIM
                           + (size_t)(k0 + koff) * NDIM + ncol;

    // per-lane prefetch base: lanes jointly cover weight rows {k..k+3}
    // (koff in {0,2}, +(lane&1)) at this tile's 64B column window.
    const float* pf = Wh + (size_t)head * KDIM * NDIM
                         + (size_t)(k0 + koff + (lane & 1)) * NDIM + ncol;

    // prime the first 32 rows of this chunk into GL2
#pragma unroll
    for (int p = 0; p < 8; ++p) {
        __builtin_prefetch(pf + (size_t)(p * 4) * NDIM, 0, 0);
    }
    pf += (size_t)32 * NDIM;   // stream-ahead distance: 32 rows (16 KB of Wh)

    v8f acc0 = {};
    v8f acc1 = {};

    // main loop: 8 k-steps per iteration, alternating accumulator chains,
    // one prefetch per 2 k-steps, single pointer bump per pointer per group
    for (int it = 0; it < KSTEPS_MAIN / 8; ++it) {
        GEMM_STEP(acc0, 0);
        GEMM_STEP(acc1, 1);
        __builtin_prefetch(pf, 0, 0);
        GEMM_STEP(acc0, 2);
        GEMM_STEP(acc1, 3);
        __builtin_prefetch(pf + 8 * NDIM, 0, 0);
        GEMM_STEP(acc0, 4);
        GEMM_STEP(acc1, 5);
        __builtin_prefetch(pf + 16 * NDIM, 0, 0);
        GEMM_STEP(acc0, 6);
        GEMM_STEP(acc1, 7);
        __builtin_prefetch(pf + 24 * NDIM, 0, 0);
        aPtr += 8 * 4;
        bPtr += 8 * 4 * NDIM;
        pf   += 8 * 4 * NDIM;
    }
    // tail: last 8 k-steps, no prefetch (stays inside this head's Wh)
    {
        GEMM_STEP(acc0, 0);
        GEMM_STEP(acc1, 1);
        GEMM_STEP(acc0, 2);
        GEMM_STEP(acc1, 3);
        GEMM_STEP(acc0, 4);
        GEMM_STEP(acc1, 5);
        GEMM_STEP(acc0, 6);
        GEMM_STEP(acc1, 7);
    }

    // merge the two accumulator chains (identical C/D lane layout)
    v8f acc = acc0 + acc1;

    // D layout (16x16 f32): VGPR v, lanes 0-15 -> M=v ; lanes 16-31 -> M=v+8
    const int mBase = (lane >> 4) << 3;
    const int nOut  = (ntile << 4) + (lane & 15);
    float* pp = partial + ((size_t)(chunk * NHEAD + head) * BATCH) * NDIM;
#pragma unroll
    for (int vg = 0; vg < 8; ++vg) {
        pp[(size_t)(mBase + vg) * NDIM + nOut] = acc[vg];
    }
}

// ---------------------------------------------------------------------------
// Kernel 3: reduce split-K partials + bias  ->  hid [7][16][128]
// ---------------------------------------------------------------------------
__global__ void __launch_bounds__(256)
reduce_bias_kernel(const float* __restrict__ partial,
                   const float* __restrict__ bh,
                   float* __restrict__ hid)
{
    const int TOT = NHEAD * BATCH * NDIM;   // 14336
    int t = blockIdx.x * blockDim.x + threadIdx.x;
    if (t >= TOT) return;

    const int n    = t & (NDIM - 1);
    const int head = t / (BATCH * NDIM);

    float s = bh[head * NDIM + n];
#pragma unroll 8
    for (int c = 0; c < NCHUNK; ++c) {
        s += partial[(size_t)c * (NHEAD * BATCH * NDIM) + t];
    }
    hid[t] = s;
}

// ---------------------------------------------------------------------------
// Kernel 4: output heads + concat  ->  out [16][242]
//   [0:4] box_xywh | [4:42] hid0@Wp+bp | [42:67] hid1@Wa+ba | [67:242] hid2..6@Wd+bd
// ---------------------------------------------------------------------------
__global__ void __launch_bounds__(256)
heads_kernel(const float* __restrict__ hid,
             const float* __restrict__ box,
             const float* __restrict__ Wp, const float* __restrict__ bp,
             const float* __restrict__ Wa, const float* __restrict__ ba,
             const float* __restrict__ Wd, const float* __restrict__ bd,
             float* __restrict__ out)
{
    const int TOT = BATCH * 242;
    int t = blockIdx.x * blockDim.x + threadIdx.x;
    if (t >= TOT) return;

    const int b   = t / 242;
    const int col = t % 242;

    float v;
    if (col < 4) {
        v = box[b * 4 + col];
    } else if (col < 42) {
        const int o = col - 4;
        const float* h = hid + 0 * (BATCH * NDIM) + b * NDIM;
        float s = bp[o];
#pragma unroll 8
        for (int q = 0; q < NDIM; ++q) s = fmaf(h[q], Wp[q * 38 + o], s);
        v = s;
    } else if (col < 67) {
        const int o = col - 42;
        const float* h = hid + 1 * (BATCH * NDIM) + b * NDIM;
        float s = ba[o];
#pragma unroll 8
        for (int q = 0; q < NDIM; ++q) s = fmaf(h[q], Wa[q * 25 + o], s);
        v = s;
    } else {
        const int oo = col - 67;
        const int kk = oo / 35;
        const int o  = oo % 35;
        const float* h = hid + (2 + kk) * (BATCH * NDIM) + b * NDIM;
        const float* w = Wd + (size_t)kk * NDIM * 35;
        float s = bd[kk * 35 + o];
#pragma unroll 8
        for (int q = 0; q < NDIM; ++q) s = fmaf(h[q], w[q * 35 + o], s);
        v = s;
    }
    out[t] = v;
}

// ---------------------------------------------------------------------------
extern "C" void kernel_launch(void* const* d_in, const int* in_sizes, int n_in,
                              void* d_out, int out_size, void* d_ws, size_t ws_size,
                              hipStream_t stream) {
    const float* box = (const float*)d_in[0];   // [16,4]
    const float* x2  = (const float*)d_in[1];   // [16,64,128,128]
    const float* x4  = (const float*)d_in[2];   // [16,160,64,64]
    const float* x6  = (const float*)d_in[3];   // [16,192,32,32]
    const float* Wh  = (const float*)d_in[4];   // [7,53248,128]
    const float* bh  = (const float*)d_in[5];   // [7,128]
    const float* Wp  = (const float*)d_in[6];   // [128,38]
    const float* bp  = (const float*)d_in[7];   // [38]
    const float* Wa  = (const float*)d_in[8];   // [128,25]
    const float* ba  = (const float*)d_in[9];   // [25]
    const float* Wd  = (const float*)d_in[10];  // [5,128,35]
    const float* bd  = (const float*)d_in[11];  // [5,35]
    float* out = (float*)d_out;                 // [16,242]

    // workspace carve-up (all 256B aligned)
    char* ws = (char*)d_ws;
    float* rois    = (float*)(ws);                       // 16*53248*4  = 3,407,872 B
    float* partial = (float*)(ws + 3407872);             // 32*7*16*128*4 = 1,835,008 B
    float* hid     = (float*)(ws + 3407872 + 1835008);   // 7*16*128*4  = 57,344 B

    // 1) ROI pooling -> rois
    {
        const int TOT = BATCH * 416 * 128;
        roi_pool_kernel<<<(TOT + 255) / 256, 256, 0, stream>>>(box, x2, x4, x6, rois);
    }
    // 2) split-K WMMA GEMM -> partial
    {
        const int waves = NCHUNK * NHEAD * 8;   // 1792 waves
        gemm_splitk_wmma<<<waves / 4, 128, 0, stream>>>(rois, Wh, partial);
    }
    // 3) reduce + bias -> hid
    {
        const int TOT = NHEAD * BATCH * NDIM;
        reduce_bias_kernel<<<(TOT + 255) / 256, 256, 0, stream>>>(partial, bh, hid);
    }
    // 4) heads + concat -> out
    {
        const int TOT = BATCH * 242;
        heads_kernel<<<(TOT + 255) / 256, 256, 0, stream>>>(hid, box, Wp, bp, Wa, ba, Wd, bd, out);
    }
}